// SumPooling_9234179686674
// MI455X (gfx1250) — compile-verified
//
#include <hip/hip_runtime.h>
#include <stdint.h>

// ---------------------------------------------------------------------------
// Segment-sum (SumPooling fwd): out[seg] += x[row] for sorted int64 index.
// Memory-bound (~1.08 GB traffic -> ~46us @ 23.3 TB/s). The segmented add is
// D = A(one-hot 16x4) x B(4x16 of x) + C via V_WMMA_F32_16X16X4_F32, so the
// accumulation runs on the matrix pipe while b128 loads stream x once.
// Column mapping: WMMA col-group c, lane-N n  <->  x column 4*n + c, which
// makes each lane's 4 col-group elements a contiguous float4 (one b128 load).
// ---------------------------------------------------------------------------

typedef __attribute__((ext_vector_type(2))) float v2f;
typedef __attribute__((ext_vector_type(8))) float v8f;

#define COLS 64          // row width (reference: x is (N, 64))
#define CHUNK 32         // rows per wave32 chunk (1 row per lane for the scan)

__global__ __launch_bounds__(256) void zero_f32(float* __restrict__ out, int n) {
    int i = blockIdx.x * blockDim.x + threadIdx.x;
    if (i < n) out[i] = 0.0f;
}

// Handles nrows % 32 leftover rows (not hit for the 4M-row reference shape).
__global__ __launch_bounds__(256) void segsum_tail(
    const float* __restrict__ x, const long long* __restrict__ idx,
    float* __restrict__ out, int start, int nrows)
{
    int t = blockIdx.x * blockDim.x + threadIdx.x;
    int total = (nrows - start) * COLS;
    if (t < total) {
        int row = start + t / COLS;
        int col = t % COLS;
        atomicAdd(out + (size_t)((int)idx[row]) * COLS + col,
                  x[(size_t)row * COLS + col]);
    }
}

__global__ __launch_bounds__(256) void segsum_wmma(
    const float* __restrict__ x,
    const long long* __restrict__ idx,
    float* __restrict__ out,
    int nchunks)
{
    const int lane = threadIdx.x & 31;
    const int wave = blockIdx.x * (blockDim.x >> 5) + (threadIdx.x >> 5);
    if (wave >= nchunks) return;                 // wave-uniform exit (EXEC stays full)

    const int row0 = wave * CHUNK;

    // ---- per-lane segment id for row (row0 + lane); all chunks are full ----
    int myseg = (int)idx[row0 + lane];

    // run-boundary flag + inclusive scan -> local slot (distinct-segment rank)
    int prev = __shfl_up(myseg, 1, 32);
    int slot = (lane > 0 && myseg != prev) ? 1 : 0;
    #pragma unroll
    for (int d = 1; d < 32; d <<= 1) {
        int t = __shfl_up(slot, d, 32);
        if (lane >= d) slot += t;
    }
    const int nslots = __shfl(slot, 31, 32) + 1; // 1..32 distinct segments

    const int half = lane >> 4;                  // lane half selects K (A/B) / M (D)
    const int nlo  = lane & 15;                  // WMMA N / (A's M) for this lane
    const int k0   = 2 * half;                   // K carried by VGPR v=0 in this lane

    // Per-lane load base: row (row0 + k0), contiguous 4 floats at col 4*nlo.
    // g advances 4 rows = 256 floats = 1024 bytes -> immediate offsets.
    const float* base0 = x + (size_t)(row0 + k0) * COLS + 4 * nlo;
    const float* base1 = base0 + COLS;           // K = k0 + 1 (VGPR v=1)

    // Process slots in windows of 16 (accumulator M dimension). Almost always
    // a single window: mean distinct segments per 32 rows is ~1.8.
    for (int base = 0; base < nslots; base += 16) {
        v8f acc0 = {0,0,0,0,0,0,0,0};
        v8f acc1 = {0,0,0,0,0,0,0,0};
        v8f acc2 = {0,0,0,0,0,0,0,0};
        v8f acc3 = {0,0,0,0,0,0,0,0};

        #pragma unroll
        for (int g = 0; g < 8; ++g) {            // K-groups: rows 4g..4g+3
            // ISA 32-bit A 16x4 layout: VGPR v, lane L -> M=L%16, K=v+2*(L/16)
            int s0 = __shfl(slot, 4 * g + k0, 32);
            int s1 = __shfl(slot, 4 * g + k0 + 1, 32);
            v2f a;
            a.x = (s0 - base == nlo) ? 1.0f : 0.0f;
            a.y = (s1 - base == nlo) ? 1.0f : 0.0f;

            // One b128 per K-row: q_v[c] = x[row0+4g+k0+v][4*nlo + c]
            float4 q0 = *(const float4*)(base0 + g * 4 * COLS);
            float4 q1 = *(const float4*)(base1 + g * 4 * COLS);

            v2f b0, b1, b2, b3;
            b0.x = q0.x;  b0.y = q1.x;
            b1.x = q0.y;  b1.y = q1.y;
            b2.x = q0.z;  b2.y = q1.z;
            b3.x = q0.w;  b3.y = q1.w;

            // D = A x B + C on the matrix pipe (f32, 16x16x4)
            acc0 = __builtin_amdgcn_wmma_f32_16x16x4_f32(false, a, false, b0,
                                                         (short)0, acc0, false, false);
            acc1 = __builtin_amdgcn_wmma_f32_16x16x4_f32(false, a, false, b1,
                                                         (short)0, acc1, false, false);
            acc2 = __builtin_amdgcn_wmma_f32_16x16x4_f32(false, a, false, b2,
                                                         (short)0, acc2, false, false);
            acc3 = __builtin_amdgcn_wmma_f32_16x16x4_f32(false, a, false, b3,
                                                         (short)0, acc3, false, false);
        }

        // ---- flush live slots: D layout: VGPR j, lanes0-15 -> M=j, lanes16-31 -> M=j+8
        const int nwin = min(nslots - base, 16);
        #pragma unroll
        for (int j = 0; j < 8; ++j) {
            #pragma unroll
            for (int h = 0; h < 2; ++h) {
                const int m = j + 8 * h;
                if (m < nwin) {                  // wave-uniform condition
                    // leader lane = first row of this slot -> its segment id
                    unsigned long long bal = __ballot(slot == base + m);
                    int leader = __ffsll(bal) - 1;
                    int seg = __shfl(myseg, leader, 32);
                    if (half == h) {             // owning half-wave holds M=m
                        float* o = out + (size_t)seg * COLS + 4 * nlo;
                        atomicAdd(o + 0, acc0[j]);
                        atomicAdd(o + 1, acc1[j]);
                        atomicAdd(o + 2, acc2[j]);
                        atomicAdd(o + 3, acc3[j]);
                    }
                }
            }
        }
    }
}

extern "C" void kernel_launch(void* const* d_in, const int* in_sizes, int n_in,
                              void* d_out, int out_size, void* d_ws, size_t ws_size,
                              hipStream_t stream) {
    const float*     x   = (const float*)d_in[0];
    const long long* idx = (const long long*)d_in[1];   // reference: int64 index
    float*           out = (float*)d_out;

    const int nrows = in_sizes[1];                      // 4,000,000 rows

    // d_out is poisoned by the harness -> zero it first (same stream = ordered)
    zero_f32<<<(out_size + 255) / 256, 256, 0, stream>>>(out, out_size);

    const int nchunks = nrows / CHUNK;                  // full 32-row chunks, no guards
    if (nchunks > 0) {
        const int wavesPerBlock = 256 / 32;             // 8 waves per 256-thread block
        const int blocks = (nchunks + wavesPerBlock - 1) / wavesPerBlock;
        segsum_wmma<<<blocks, 256, 0, stream>>>(x, idx, out, nchunks);
    }

    const int tail_start = nchunks * CHUNK;             // leftover rows (0 for 4M)
    const int tail_rows  = nrows - tail_start;
    if (tail_rows > 0) {
        const int tail_elems = tail_rows * COLS;
        segsum_tail<<<(tail_elems + 255) / 256, 256, 0, stream>>>(
            x, idx, out, tail_start, nrows);
    }
}